// GaussRenderer_24696061952307
// MI455X (gfx1250) — compile-verified
//
#include <hip/hip_runtime.h>
#include <math.h>

#define NG     1024
#define IMH    128
#define IMW    128
#define TS     32
#define FOCAL_ 1.6f
#define CHUNK  16
#define MSTRIDE 17

typedef __attribute__((ext_vector_type(2))) float v2f;
typedef __attribute__((ext_vector_type(8))) float v8f;

// workspace layout (float offsets)
enum {
  OFF_DEPTH_U = 0,
  OFF_COEFF_U = 1024,
  OFF_OP_U    = 9216,
  OFF_COL_U   = 10240,
  OFF_BBOX_U  = 13312,
  OFF_DEPTH_S = 17408,
  OFF_COEFF_S = 18432,
  OFF_OP_S    = 26624,
  OFF_COL_S   = 27648,
  OFF_BBOX_S  = 30720,
  WS_FLOATS   = 34816
};

// ---------------------------------------------------------------------------
// Kernel 1: per-gaussian preprocess (projection, SH color, cov2d -> 8 mahal
// coefficients in pixel space, bbox, radii)
// ---------------------------------------------------------------------------
__global__ void gs_preprocess(const float* __restrict__ xyz,
                              const float* __restrict__ feat,
                              const float* __restrict__ scal,
                              const float* __restrict__ rot,
                              const float* __restrict__ opac,
                              const float* __restrict__ Vm,
                              const float* __restrict__ Pm,
                              const float* __restrict__ cam,
                              float* __restrict__ ws,
                              float* __restrict__ radii_out)
{
  int n = blockIdx.x * blockDim.x + threadIdx.x;
  if (n >= NG) return;

  float px = xyz[n*3+0], py = xyz[n*3+1], pz = xyz[n*3+2];

  // p_view = [p,1] @ V ; p_h = p_view @ P
  float pv[4], ph[4];
  #pragma unroll
  for (int j = 0; j < 4; ++j)
    pv[j] = px*Vm[0*4+j] + py*Vm[1*4+j] + pz*Vm[2*4+j] + Vm[3*4+j];
  #pragma unroll
  for (int j = 0; j < 4; ++j)
    ph[j] = pv[0]*Pm[0*4+j] + pv[1]*Pm[1*4+j] + pv[2]*Pm[2*4+j] + pv[3]*Pm[3*4+j];
  float invw  = 1.0f / (ph[3] + 1e-6f);
  float projx = ph[0]*invw, projy = ph[1]*invw;
  float depth = pv[2];

  // ---- SH (deg 3), dirs = xyz - camera_center (unnormalized, as reference)
  float rdx = px - cam[0], rdy = py - cam[1], rdz = pz - cam[2];
  float xx = rdx*rdx, yy = rdy*rdy, zz = rdz*rdz;
  float xy = rdx*rdy, yz = rdy*rdz, xz = rdx*rdz;
  float col[3];
  const float* F = feat + n*48;  // (16,3) row-major per gaussian
  #pragma unroll
  for (int c = 0; c < 3; ++c) {
    float s[16];
    #pragma unroll
    for (int k = 0; k < 16; ++k) s[k] = F[k*3 + c];
    float res = 0.28209479177387814f*s[0];
    res += -0.4886025119029199f*rdy*s[1] + 0.4886025119029199f*rdz*s[2]
         -  0.4886025119029199f*rdx*s[3];
    res +=  1.0925484305920792f*xy*s[4] - 1.0925484305920792f*yz*s[5]
         +  0.31539156525252005f*(2.0f*zz - xx - yy)*s[6]
         -  1.0925484305920792f*xz*s[7] + 0.5462742152960396f*(xx - yy)*s[8];
    res += -0.5900435899266435f*rdy*(3.0f*xx - yy)*s[9]
         +  2.890611442640554f*xy*rdz*s[10]
         -  0.4570457994644658f*rdy*(4.0f*zz - xx - yy)*s[11]
         +  0.3731763325901154f*rdz*(2.0f*zz - 3.0f*xx - 3.0f*yy)*s[12]
         -  0.4570457994644658f*rdx*(4.0f*zz - xx - yy)*s[13]
         +  1.445305721320277f*rdz*(xx - yy)*s[14]
         -  0.5900435899266435f*rdx*(xx - 3.0f*yy)*s[15];
    col[c] = fmaxf(res + 0.5f, 0.0f);
  }

  // ---- cov3d = (R*S)(R*S)^T
  float qw = rot[n*4+0], qx = rot[n*4+1], qy = rot[n*4+2], qz = rot[n*4+3];
  float qn = rsqrtf(qw*qw + qx*qx + qy*qy + qz*qz);
  qw *= qn; qx *= qn; qy *= qn; qz *= qn;
  float R[3][3] = {
    {1.0f-2.0f*(qy*qy+qz*qz), 2.0f*(qx*qy-qw*qz),      2.0f*(qx*qz+qw*qy)},
    {2.0f*(qx*qy+qw*qz),      1.0f-2.0f*(qx*qx+qz*qz), 2.0f*(qy*qz-qw*qx)},
    {2.0f*(qx*qz-qw*qy),      2.0f*(qy*qz+qw*qx),      1.0f-2.0f*(qx*qx+qy*qy)}
  };
  float sc[3] = { scal[n*3+0], scal[n*3+1], scal[n*3+2] };
  float L[3][3], cov3[3][3];
  #pragma unroll
  for (int i = 0; i < 3; ++i)
    #pragma unroll
    for (int j = 0; j < 3; ++j) L[i][j] = R[i][j]*sc[j];
  #pragma unroll
  for (int i = 0; i < 3; ++i)
    #pragma unroll
    for (int j = 0; j < 3; ++j)
      cov3[i][j] = L[i][0]*L[j][0] + L[i][1]*L[j][1] + L[i][2]*L[j][2];

  // ---- project covariance: J (W^T cov3 W) J^T
  float tz   = pv[2];
  float tanf_ = tanf(0.5f * FOCAL_);
  float lim  = 1.3f * tanf_;
  float txv = fminf(fmaxf(pv[0]/tz, -lim), lim) * tz;
  float tyv = fminf(fmaxf(pv[1]/tz, -lim), lim) * tz;
  float fz  = FOCAL_ / tz;
  float j02 = -txv * FOCAL_ / (tz*tz);
  float j12 = -tyv * FOCAL_ / (tz*tz);

  float Wt[3][3];   // Wm = view[:3,:3]^T
  #pragma unroll
  for (int i = 0; i < 3; ++i)
    #pragma unroll
    for (int j = 0; j < 3; ++j) Wt[i][j] = Vm[j*4 + i];
  float t1[3][3], M[3][3];
  #pragma unroll
  for (int j = 0; j < 3; ++j)
    #pragma unroll
    for (int l = 0; l < 3; ++l)
      t1[j][l] = cov3[j][0]*Wt[l][0] + cov3[j][1]*Wt[l][1] + cov3[j][2]*Wt[l][2];
  #pragma unroll
  for (int i = 0; i < 3; ++i)
    #pragma unroll
    for (int l = 0; l < 3; ++l)
      M[i][l] = Wt[i][0]*t1[0][l] + Wt[i][1]*t1[1][l] + Wt[i][2]*t1[2][l];

  float u0[3], u1[3];
  #pragma unroll
  for (int k = 0; k < 3; ++k) {
    u0[k] = fz*M[0][k] + j02*M[2][k];
    u1[k] = fz*M[1][k] + j12*M[2][k];
  }
  float a  = u0[0]*fz + u0[2]*j02 + 0.3f;
  float b  = u0[1]*fz + u0[2]*j12;
  float cc = u1[0]*fz + u1[2]*j02;
  float d  = u1[1]*fz + u1[2]*j12 + 0.3f;

  float det = a*d - b*cc;
  float mid = 0.5f*(a + d);
  float sq  = sqrtf(fmaxf(mid*mid - det, 0.1f));
  float lam = fmaxf(mid + sq, mid - sq);
  float radii = 3.0f * ceilf(sqrtf(lam));

  float m0 = ((projx + 1.0f)*IMW - 1.0f)*0.5f;
  float m1 = ((projy + 1.0f)*IMH - 1.0f)*0.5f;
  float rminx = fminf(fmaxf(m0 - radii, 0.0f), IMW - 1.0f);
  float rminy = fminf(fmaxf(m1 - radii, 0.0f), IMH - 1.0f);
  float rmaxx = fminf(fmaxf(m0 + radii, 0.0f), IMW - 1.0f);
  float rmaxy = fminf(fmaxf(m1 + radii, 0.0f), IMH - 1.0f);

  // inv_cov and 8 mahal coefficients:
  //   mahal(u,v) = ia*du^2 + s*(du*dv) + id*dv^2, du=u-m0, dv=v-m1
  //   features per pixel: [u^2, u, v^2, v, u*v, 1, 0, 0]
  float inv_det = 1.0f / det;
  float ia  = d  * inv_det;
  float ib  = -b * inv_det;
  float ic  = -cc* inv_det;
  float idd = a  * inv_det;
  float s2  = ib + ic;
  float k0 = ia;
  float k1 = -2.0f*ia*m0 - s2*m1;
  float k2 = idd;
  float k3 = -2.0f*idd*m1 - s2*m0;
  float k4 = s2;
  float k5 = ia*m0*m0 + s2*m0*m1 + idd*m1*m1;

  ws[OFF_DEPTH_U + n] = depth;
  ws[OFF_OP_U    + n] = opac[n];
  ws[OFF_COL_U + n*3 + 0] = col[0];
  ws[OFF_COL_U + n*3 + 1] = col[1];
  ws[OFF_COL_U + n*3 + 2] = col[2];
  ws[OFF_BBOX_U + n*4 + 0] = rminx;
  ws[OFF_BBOX_U + n*4 + 1] = rminy;
  ws[OFF_BBOX_U + n*4 + 2] = rmaxx;
  ws[OFF_BBOX_U + n*4 + 3] = rmaxy;
  float* cf = ws + OFF_COEFF_U + n*8;
  cf[0]=k0; cf[1]=k1; cf[2]=k2; cf[3]=k3; cf[4]=k4; cf[5]=k5; cf[6]=0.0f; cf[7]=0.0f;

  radii_out[n] = radii;
}

// ---------------------------------------------------------------------------
// Kernel 2: single-block bitonic argsort over depth (N=1024) + gather into
// depth-sorted SoA arrays
// ---------------------------------------------------------------------------
__global__ void gs_sort_gather(float* __restrict__ ws)
{
  __shared__ float key[NG];
  __shared__ int   idx[NG];
  int tid = threadIdx.x;
  key[tid] = ws[OFF_DEPTH_U + tid];
  idx[tid] = tid;
  __syncthreads();
  for (int k = 2; k <= NG; k <<= 1) {
    for (int j = k >> 1; j > 0; j >>= 1) {
      int ixj = tid ^ j;
      if (ixj > tid) {
        float ka = key[tid], kb = key[ixj];
        bool up = ((tid & k) == 0);
        if ((ka > kb) == up) {
          key[tid] = kb; key[ixj] = ka;
          int t = idx[tid]; idx[tid] = idx[ixj]; idx[ixj] = t;
        }
      }
      __syncthreads();
    }
  }
  int g = idx[tid];
  ws[OFF_DEPTH_S + tid] = ws[OFF_DEPTH_U + g];
  ws[OFF_OP_S    + tid] = ws[OFF_OP_U    + g];
  #pragma unroll
  for (int q = 0; q < 8; ++q)
    ws[OFF_COEFF_S + tid*8 + q] = ws[OFF_COEFF_U + g*8 + q];
  #pragma unroll
  for (int q = 0; q < 3; ++q)
    ws[OFF_COL_S + tid*3 + q] = ws[OFF_COL_U + g*3 + q];
  #pragma unroll
  for (int q = 0; q < 4; ++q)
    ws[OFF_BBOX_S + tid*4 + q] = ws[OFF_BBOX_U + g*4 + q];
}

// ---------------------------------------------------------------------------
// Kernel 3: tiled render. 64 blocks x 256 threads: block = quarter of a
// 32x32 tile (256 pixels). Mahalanobis matrix (256 px x 16 gaussians) per
// chunk is computed with FP32 WMMA (two chained 16x16x4 ops, K padded 6->8),
// staged through LDS, then each thread alpha-composites its own pixel.
// ---------------------------------------------------------------------------
__global__ void __launch_bounds__(256)
gs_render(const float* __restrict__ ws, float* __restrict__ out)
{
  __shared__ float s_mahal[256 * MSTRIDE];
  __shared__ float s_op[CHUNK];
  __shared__ float s_dep[CHUNK];
  __shared__ float s_col[CHUNK*3];
  __shared__ int   s_mask[CHUNK];
  __shared__ int   s_chunkAny;
  __shared__ int   s_tileAny;

  const int tid  = threadIdx.x;
  const int blk  = blockIdx.x;
  const int tile = blk >> 2;       // 16 tiles
  const int sub  = blk & 3;        // quarter-tile within tile
  const int th   = tile >> 2, tw = tile & 3;   // nTh = nTw = 4
  const int h0i  = th * TS, w0i = tw * TS;
  const float h0 = (float)h0i, w0 = (float)w0i;

  const int pt = sub*256 + tid;    // pixel index within tile (0..1023)
  const int pa = pt >> 5, pb = pt & 31;

  const int lane  = tid & 31;
  const int wave  = tid >> 5;      // 8 waves
  const int m     = lane & 15;
  const int khalf = (lane >> 4) << 1;     // A/B half-K select per ISA layout

  // A operands (pixel features), invariant across chunks. Wave handles two
  // 16-pixel groups: pg = 2*wave, 2*wave+1.
  float fA[2][8];
  #pragma unroll
  for (int t2 = 0; t2 < 2; ++t2) {
    int p   = (wave*2 + t2)*16 + m;   // pixel-in-block of row M=m
    int ptl = sub*256 + p;            // pixel-in-tile
    float uu = (float)(h0i + (ptl >> 5));
    float vv = (float)(w0i + (ptl & 31));
    fA[t2][0] = uu*uu; fA[t2][1] = uu;
    fA[t2][2] = vv*vv; fA[t2][3] = vv;
    fA[t2][4] = uu*vv; fA[t2][5] = 1.0f;
    fA[t2][6] = 0.0f;  fA[t2][7] = 0.0f;
  }
  v2f a00 = { fA[0][khalf],   fA[0][khalf+1] };
  v2f a01 = { fA[0][khalf+4], fA[0][khalf+5] };
  v2f a10 = { fA[1][khalf],   fA[1][khalf+1] };
  v2f a11 = { fA[1][khalf+4], fA[1][khalf+5] };

  if (tid == 0) s_tileAny = 0;

  const float* coeff_s = ws + OFF_COEFF_S;
  float T = 1.0f;
  float accR = 0.0f, accG = 0.0f, accB = 0.0f, accD = 0.0f, accA = 0.0f;

  for (int ch = 0; ch < NG / CHUNK; ++ch) {
    if (tid == 0) s_chunkAny = 0;
    __syncthreads();   // orders previous composite before meta overwrite

    if (tid < CHUNK) {
      int g = ch*CHUNK + tid;
      float rminx = ws[OFF_BBOX_S + g*4 + 0];
      float rminy = ws[OFF_BBOX_S + g*4 + 1];
      float rmaxx = ws[OFF_BBOX_S + g*4 + 2];
      float rmaxy = ws[OFF_BBOX_S + g*4 + 3];
      int mk = (rmaxx >= w0) && (rminx <= w0 + (float)(TS-1)) &&
               (rmaxy >= h0) && (rminy <= h0 + (float)(TS-1));
      s_mask[tid] = mk;
      s_op[tid]   = ws[OFF_OP_S + g];
      s_dep[tid]  = ws[OFF_DEPTH_S + g];
      s_col[tid*3+0] = ws[OFF_COL_S + g*3 + 0];
      s_col[tid*3+1] = ws[OFF_COL_S + g*3 + 1];
      s_col[tid*3+2] = ws[OFF_COL_S + g*3 + 2];
      if (mk) { s_chunkAny = 1; s_tileAny = 1; }
    }
    if (ch + 1 < NG / CHUNK)
      __builtin_prefetch(coeff_s + (ch + 1)*CHUNK*8, 0, 1);
    __syncthreads();

    const int active = s_chunkAny;
    if (active) {
      // B operand: coefficients of gaussian column n = lane&15
      int g = ch*CHUNK + m;
      v2f b0 = { coeff_s[g*8 + khalf],     coeff_s[g*8 + khalf + 1] };
      v2f b1 = { coeff_s[g*8 + khalf + 4], coeff_s[g*8 + khalf + 5] };

      v8f acc0 = {0.f,0.f,0.f,0.f,0.f,0.f,0.f,0.f};
      acc0 = __builtin_amdgcn_wmma_f32_16x16x4_f32(false, a00, false, b0,
                                                   (short)0, acc0, false, false);
      acc0 = __builtin_amdgcn_wmma_f32_16x16x4_f32(false, a01, false, b1,
                                                   (short)0, acc0, false, false);
      v8f acc1 = {0.f,0.f,0.f,0.f,0.f,0.f,0.f,0.f};
      acc1 = __builtin_amdgcn_wmma_f32_16x16x4_f32(false, a10, false, b0,
                                                   (short)0, acc1, false, false);
      acc1 = __builtin_amdgcn_wmma_f32_16x16x4_f32(false, a11, false, b1,
                                                   (short)0, acc1, false, false);

      // D layout: VGPR r -> row M=(lane>>4)*8+r, col N=lane&15
      int rowbase = (lane >> 4) * 8;
      int pg0 = wave*2, pg1 = wave*2 + 1;
      #pragma unroll
      for (int r = 0; r < 8; ++r) {
        s_mahal[(pg0*16 + rowbase + r)*MSTRIDE + m] = acc0[r];
        s_mahal[(pg1*16 + rowbase + r)*MSTRIDE + m] = acc1[r];
      }
    }
    __syncthreads();

    if (active) {
      #pragma unroll
      for (int gg = 0; gg < CHUNK; ++gg) {
        if (s_mask[gg]) {
          float mh    = s_mahal[tid*MSTRIDE + gg];
          float alpha = s_op[gg] * __expf(-0.5f * mh);
          float wgt   = T * alpha;
          accR += wgt * s_col[gg*3 + 0];
          accG += wgt * s_col[gg*3 + 1];
          accB += wgt * s_col[gg*3 + 2];
          accD += wgt * s_dep[gg];
          accA += wgt;
          T *= (1.0f - alpha + 1e-10f);
        }
      }
    }
  }
  __syncthreads();

  const int anyT = s_tileAny;
  const int gh = h0i + pa, gw = w0i + pb;
  const int pix = gh*IMW + gw;
  out[pix*3 + 0] = anyT ? accR : 1.0f;
  out[pix*3 + 1] = anyT ? accG : 1.0f;
  out[pix*3 + 2] = anyT ? accB : 1.0f;
  out[IMH*IMW*3 + pix] = accD;
  out[IMH*IMW*4 + pix] = accA;
}

// ---------------------------------------------------------------------------
extern "C" void kernel_launch(void* const* d_in, const int* in_sizes, int n_in,
                              void* d_out, int out_size, void* d_ws, size_t ws_size,
                              hipStream_t stream)
{
  const float* xyz  = (const float*)d_in[0];
  const float* feat = (const float*)d_in[1];
  const float* scal = (const float*)d_in[2];
  const float* rot  = (const float*)d_in[3];
  const float* opac = (const float*)d_in[4];
  const float* Vm   = (const float*)d_in[5];
  const float* Pm   = (const float*)d_in[6];
  const float* cam  = (const float*)d_in[7];
  float* out = (float*)d_out;
  float* ws  = (float*)d_ws;

  float* radii_out = out + IMH*IMW*5;   // after col(3)+dep(1)+alpha(1)

  gs_preprocess<<<NG/256, 256, 0, stream>>>(xyz, feat, scal, rot, opac,
                                            Vm, Pm, cam, ws, radii_out);
  gs_sort_gather<<<1, NG, 0, stream>>>(ws);
  gs_render<<<64, 256, 0, stream>>>(ws, out);
}